// Sparsify_80204219286180
// MI455X (gfx1250) — compile-verified
//
#include <hip/hip_runtime.h>
#include <hip/hip_bf16.h>

// Per-8-element-block top-4-by-|x| mask (stable-argsort tie break), x * mask.
// Pure HBM-streaming kernel: 2x global_load_b128 (NT) + ~100 VALU + 2x
// global_store_b128 (NT) per lane; one 8-float block per lane.

typedef __attribute__((ext_vector_type(4))) float v4f;

__global__ __launch_bounds__(256) void
sparsify_blk8_top4(const float* __restrict__ x, float* __restrict__ y, int nblocks) {
    int b = blockIdx.x * blockDim.x + threadIdx.x;
    if (b >= nblocks) return;

    const v4f* __restrict__ px = reinterpret_cast<const v4f*>(x) + (size_t)b * 2;
    v4f*       __restrict__ py = reinterpret_cast<v4f*>(y)       + (size_t)b * 2;

    // Two B128 non-temporal loads: 8 contiguous floats = one block.
    v4f v0 = __builtin_nontemporal_load(px);
    v4f v1 = __builtin_nontemporal_load(px + 1);

    float v[8] = {v0.x, v0.y, v0.z, v0.w, v1.x, v1.y, v1.z, v1.w};

    float a[8];
#pragma unroll
    for (int i = 0; i < 8; ++i) a[i] = __builtin_fabsf(v[i]);

    // cnt[i] = number of elements that outrank element i under the stable
    // argsort order: j outranks i iff a[j] > a[i], or a[j] == a[i] && j > i.
    // For each pair (i<j) exactly one of (a[j] >= a[i]) / (a[i] > a[j]) holds.
    int cnt[8] = {0, 0, 0, 0, 0, 0, 0, 0};
#pragma unroll
    for (int i = 0; i < 8; ++i) {
#pragma unroll
        for (int j = i + 1; j < 8; ++j) {
            int jwins = (a[j] >= a[i]);   // tie -> later index wins
            cnt[i] += jwins;
            cnt[j] += 1 - jwins;
        }
    }

    // Keep the 4 top-ranked elements (cnt < 4), zero the rest.
#pragma unroll
    for (int i = 0; i < 8; ++i) v[i] = (cnt[i] < 4) ? v[i] : 0.0f;

    v4f o0 = {v[0], v[1], v[2], v[3]};
    v4f o1 = {v[4], v[5], v[6], v[7]};

    // Two B128 non-temporal stores.
    __builtin_nontemporal_store(o0, py);
    __builtin_nontemporal_store(o1, py + 1);
}

extern "C" void kernel_launch(void* const* d_in, const int* in_sizes, int n_in,
                              void* d_out, int out_size, void* d_ws, size_t ws_size,
                              hipStream_t stream) {
    const float* x = (const float*)d_in[0];
    float* y = (float*)d_out;

    int n = in_sizes[0];          // 4096 * 16384 = 67,108,864 floats
    int nblocks = n / 8;          // 8,388,608 blocks of 8

    const int threads = 256;      // 8 wave32 waves per workgroup
    int groups = (nblocks + threads - 1) / threads;

    sparsify_blk8_top4<<<groups, threads, 0, stream>>>(x, y, nblocks);
}